// GCN_6545530159286
// MI455X (gfx1250) — compile-verified
//
#include <hip/hip_runtime.h>

typedef float v2f __attribute__((ext_vector_type(2)));
typedef float v8f __attribute__((ext_vector_type(8)));

static constexpr int   NODES   = 102400;          // 400 ROI * 256 graphs
static constexpr long  EDGES   = 3276800;         // N * 32
static constexpr int   T_IN    = 64;
static constexpr int   H_HID   = 32;
static constexpr int   BS_OUT  = 16;
static constexpr int   GRAPHS  = 256;
static constexpr int   FC1_IN  = 6400;            // 400 * 16
static constexpr int   FC1_OUT = 256;
static constexpr int   FC2_OUT = 128;
static constexpr int   NCLS    = 9;

// ---------------------------------------------------------------- utilities

__global__ void zero_f32(float* __restrict__ p, long n) {
  long i = (long)blockIdx.x * blockDim.x + threadIdx.x;
  if (i < n) p[i] = 0.0f;
}

__global__ void deg_accum(const int* __restrict__ src,
                          const float* __restrict__ w,
                          float* __restrict__ deg) {
  long e = (long)blockIdx.x * blockDim.x + threadIdx.x;
  if (e < EDGES) atomicAdd(&deg[src[e]], w[e]);
}

__global__ void finish_dinv(float* __restrict__ deg) {
  long i = (long)blockIdx.x * blockDim.x + threadIdx.x;
  if (i < NODES) {
    float d = deg[i];
    deg[i] = (d > 0.0f) ? rsqrtf(d) : 0.0f;   // in place: deg -> dinv
  }
}

__global__ void edge_lapw(const int* __restrict__ src,
                          const int* __restrict__ dst,
                          const float* __restrict__ w,
                          const float* __restrict__ dinv,
                          float* __restrict__ lapw) {
  long e = (long)blockIdx.x * blockDim.x + threadIdx.x;
  if (e < EDGES) lapw[e] = -(dinv[src[e]] * w[e] * dinv[dst[e]]);
}

// tx[dst] += lapw[e] * h[src]   (one thread per edge per 4 channels)
template <int C>
__global__ void scatter_edges(const int* __restrict__ src,
                              const int* __restrict__ dst,
                              const float* __restrict__ lapw,
                              const float* __restrict__ h,
                              float* __restrict__ tx) {
  constexpr int G = C / 4;
  long tid = (long)blockIdx.x * blockDim.x + threadIdx.x;
  long e  = tid / G;
  int  c4 = (int)(tid % G) * 4;
  if (e >= EDGES) return;
  float lw = lapw[e];
  const float4 v = *(const float4*)(h + (long)src[e] * C + c4);
  float* t = tx + (long)dst[e] * C + c4;
  atomicAdd(t + 0, lw * v.x);
  atomicAdd(t + 1, lw * v.y);
  atomicAdd(t + 2, lw * v.z);
  atomicAdd(t + 3, lw * v.w);
}

// transpose W[K,Ncol] -> Wt[Ncol,K] (one-shot, for vectorized B loads)
__global__ void transpose_w(const float* __restrict__ w, float* __restrict__ wt,
                            int K, int Ncol) {
  long i = (long)blockIdx.x * blockDim.x + threadIdx.x;
  if (i >= (long)K * Ncol) return;
  int k = (int)(i / Ncol), c = (int)(i % Ncol);
  wt[(long)c * K + k] = w[i];
}

// ------------------------------------------------------------- WMMA helpers
// A 16x4 f32: lanes 0-15 -> K=0,1 ; lanes 16-31 -> K=2,3  (ISA 7.12.2)
__device__ __forceinline__ v2f load_a(const float* __restrict__ A, int K,
                                      int row, int k, int half) {
  return *(const v2f*)(A + (long)row * K + k + half * 2);
}

// =====================  Cheb combine (fused, LDS weights)  =================
// out[N,NCOL] = relu( A0@W0 + A1@W1 + bias )
// Block = 8 waves; each wave owns one 16-row M-tile and ALL NCOL/16 column
// tiles (A fragments loaded once, reused). Weights staged transposed in LDS.
template <int K, int NCOL>
__global__ void cheb_combine_wmma(const float* __restrict__ A0,
                                  const float* __restrict__ W0,
                                  const float* __restrict__ A1,
                                  const float* __restrict__ W1,
                                  const float* __restrict__ bias,
                                  float* __restrict__ out) {
  constexpr int NT = NCOL / 16;
  __shared__ float w0t[NCOL * K];   // [col][k]
  __shared__ float w1t[NCOL * K];

  for (int i = threadIdx.x; i < K * NCOL; i += blockDim.x) {
    int k = i / NCOL, c = i % NCOL;
    w0t[c * K + k] = W0[i];
    w1t[c * K + k] = W1[i];
  }
  __syncthreads();

  int wave = (int)(((long)blockIdx.x * blockDim.x + threadIdx.x) >> 5); // M tile
  int lane = threadIdx.x & 31;
  int half = lane >> 4;
  int mn   = lane & 15;
  int row  = wave * 16 + mn;

  v8f c[NT];
#pragma unroll
  for (int t = 0; t < NT; ++t) {
    float b = bias[t * 16 + mn];
#pragma unroll
    for (int v = 0; v < 8; ++v) c[t][v] = b;
  }

#pragma unroll 4
  for (int k = 0; k < K; k += 4) {
    v2f a0 = load_a(A0, K, row, k, half);
    v2f a1 = load_a(A1, K, row, k, half);
#pragma unroll
    for (int t = 0; t < NT; ++t) {
      int col = t * 16 + mn;
      v2f b0 = *(const v2f*)&w0t[col * K + k + half * 2];   // ds_load_b64
      c[t] = __builtin_amdgcn_wmma_f32_16x16x4_f32(false, a0, false, b0,
                                                   (short)0, c[t], false, false);
      v2f b1 = *(const v2f*)&w1t[col * K + k + half * 2];
      c[t] = __builtin_amdgcn_wmma_f32_16x16x4_f32(false, a1, false, b1,
                                                   (short)0, c[t], false, false);
    }
  }

#pragma unroll
  for (int t = 0; t < NT; ++t)
#pragma unroll
    for (int v = 0; v < 8; ++v) {
      float val = fmaxf(c[t][v], 0.0f);                      // conv: ReLU
      out[(long)(wave * 16 + v + 8 * half) * NCOL + t * 16 + mn] = val;
    }
}

// =====================  FC GEMM (transposed weights, A reuse)  =============
// out[M,Ncol] = A@W + bias ; Wt is [Ncol,K]; each wave does NT column tiles.
template <int NT>
__global__ void gemm_bias_wmma(const float* __restrict__ A,
                               const float* __restrict__ Wt,
                               const float* __restrict__ bias,
                               float* __restrict__ out,
                               int K, int Ncol) {
  int wave = (int)(((long)blockIdx.x * blockDim.x + threadIdx.x) >> 5);
  int lane = threadIdx.x & 31;
  int half = lane >> 4;
  int mn   = lane & 15;
  int ngroups = Ncol / (16 * NT);
  int mt = wave / ngroups;
  int ng = wave % ngroups;
  int row = mt * 16 + mn;

  v8f c[NT];
#pragma unroll
  for (int t = 0; t < NT; ++t) {
    float b = bias[(ng * NT + t) * 16 + mn];
#pragma unroll
    for (int v = 0; v < 8; ++v) c[t][v] = b;
  }

#pragma unroll 4
  for (int k = 0; k < K; k += 4) {
    v2f a = load_a(A, K, row, k, half);
#pragma unroll
    for (int t = 0; t < NT; ++t) {
      int col = (ng * NT + t) * 16 + mn;
      v2f b = *(const v2f*)(Wt + (long)col * K + k + half * 2);
      c[t] = __builtin_amdgcn_wmma_f32_16x16x4_f32(false, a, false, b,
                                                   (short)0, c[t], false, false);
    }
  }

#pragma unroll
  for (int t = 0; t < NT; ++t)
#pragma unroll
    for (int v = 0; v < 8; ++v)
      out[(long)(mt * 16 + v + 8 * half) * Ncol + (ng * NT + t) * 16 + mn] = c[t][v];
}

// fc3: [256,128] @ [128,9] + b  -> [256,9]  (ragged 9 cols: scalar kernel)
__global__ void fc3_kernel(const float* __restrict__ h,
                           const float* __restrict__ W,
                           const float* __restrict__ b,
                           float* __restrict__ out) {
  int i = blockIdx.x * blockDim.x + threadIdx.x;
  if (i >= GRAPHS * NCLS) return;
  int g = i / NCLS, c = i % NCLS;
  float acc = b[c];
#pragma unroll 8
  for (int k = 0; k < FC2_OUT; ++k)
    acc += h[g * FC2_OUT + k] * W[k * NCLS + c];
  out[i] = acc;
}

// ------------------------------------------------------------------ launch

extern "C" void kernel_launch(void* const* d_in, const int* in_sizes, int n_in,
                              void* d_out, int out_size, void* d_ws, size_t ws_size,
                              hipStream_t stream) {
  const float* x    = (const float*)d_in[0];
  const int*   src  = (const int*)d_in[1];            // edge_index[0]
  const int*   dst  = ((const int*)d_in[1]) + EDGES;  // edge_index[1]
  const float* ew   = (const float*)d_in[2];
  const float* W1_0 = (const float*)d_in[3];
  const float* W1_1 = (const float*)d_in[4];
  const float* b1   = (const float*)d_in[5];
  const float* W2_0 = (const float*)d_in[6];
  const float* W2_1 = (const float*)d_in[7];
  const float* b2   = (const float*)d_in[8];
  const float* W3_0 = (const float*)d_in[9];
  const float* W3_1 = (const float*)d_in[10];
  const float* b3   = (const float*)d_in[11];
  const float* fc1w = (const float*)d_in[12];
  const float* fc1b = (const float*)d_in[13];
  const float* fc2w = (const float*)d_in[14];
  const float* fc2b = (const float*)d_in[15];
  const float* fc3w = (const float*)d_in[16];
  const float* fc3b = (const float*)d_in[17];
  float* out = (float*)d_out;

  // workspace carve-up (floats)
  float* ws   = (float*)d_ws;
  float* deg  = ws;                              // N        (becomes dinv)
  float* lapw = deg  + NODES;                    // E
  float* tx   = lapw + EDGES;                    // N*64 (reused at N*32)
  float* h1   = tx   + (long)NODES * T_IN;       // N*32
  float* h2   = h1   + (long)NODES * H_HID;      // N*32
  float* h3   = h2   + (long)NODES * H_HID;      // N*16
  float* f1   = h3   + (long)NODES * BS_OUT;     // 256*256
  float* f2   = f1   + GRAPHS * FC1_OUT;         // 256*128
  float* wt1  = f2   + GRAPHS * FC2_OUT;         // 6400*256 (fc1_w transposed)
  float* wt2  = wt1  + (long)FC1_IN * FC1_OUT;   // 256*128  (fc2_w transposed)

  const int TB = 256;
  auto blocks = [](long n, int tb) { return (unsigned)((n + tb - 1) / tb); };

  // ---- scaled Laplacian edge weights
  zero_f32<<<blocks(NODES, TB), TB, 0, stream>>>(deg, NODES);
  deg_accum<<<blocks(EDGES, TB), TB, 0, stream>>>(src, ew, deg);
  finish_dinv<<<blocks(NODES, TB), TB, 0, stream>>>(deg);
  edge_lapw<<<blocks(EDGES, TB), TB, 0, stream>>>(src, dst, ew, deg, lapw);

  // ---- one-shot weight transposes for FC layers
  transpose_w<<<blocks((long)FC1_IN * FC1_OUT, TB), TB, 0, stream>>>(
      fc1w, wt1, FC1_IN, FC1_OUT);
  transpose_w<<<blocks((long)FC1_OUT * FC2_OUT, TB), TB, 0, stream>>>(
      fc2w, wt2, FC1_OUT, FC2_OUT);

  // ---- Cheb layer 1: x[N,64] -> h1[N,32]
  zero_f32<<<blocks((long)NODES * T_IN, TB), TB, 0, stream>>>(tx, (long)NODES * T_IN);
  scatter_edges<T_IN><<<blocks(EDGES * (T_IN / 4), TB), TB, 0, stream>>>(src, dst, lapw, x, tx);
  cheb_combine_wmma<T_IN, H_HID><<<NODES / 16 / 8, TB, 0, stream>>>(
      x, W1_0, tx, W1_1, b1, h1);

  // ---- Cheb layer 2: h1[N,32] -> h2[N,32]
  zero_f32<<<blocks((long)NODES * H_HID, TB), TB, 0, stream>>>(tx, (long)NODES * H_HID);
  scatter_edges<H_HID><<<blocks(EDGES * (H_HID / 4), TB), TB, 0, stream>>>(src, dst, lapw, h1, tx);
  cheb_combine_wmma<H_HID, H_HID><<<NODES / 16 / 8, TB, 0, stream>>>(
      h1, W2_0, tx, W2_1, b2, h2);

  // ---- Cheb layer 3: h2[N,32] -> h3[N,16]
  zero_f32<<<blocks((long)NODES * H_HID, TB), TB, 0, stream>>>(tx, (long)NODES * H_HID);
  scatter_edges<H_HID><<<blocks(EDGES * (H_HID / 4), TB), TB, 0, stream>>>(src, dst, lapw, h2, tx);
  cheb_combine_wmma<H_HID, BS_OUT><<<NODES / 16 / 8, TB, 0, stream>>>(
      h2, W3_0, tx, W3_1, b3, h3);

  // ---- fc1: h3 viewed as [256, 6400] @ [6400,256] + b  (NT=4 -> 64 waves)
  gemm_bias_wmma<4><<<(GRAPHS / 16) * (FC1_OUT / 64) * 32 / TB, TB, 0, stream>>>(
      h3, wt1, fc1b, f1, FC1_IN, FC1_OUT);

  // ---- fc2: [256,256] @ [256,128] + b  (NT=4 -> 32 waves)
  gemm_bias_wmma<4><<<(GRAPHS / 16) * (FC2_OUT / 64) * 32 / TB, TB, 0, stream>>>(
      f1, wt2, fc2b, f2, FC1_OUT, FC2_OUT);

  // ---- fc3: [256,128] @ [128,9] + b -> out [256,9]
  fc3_kernel<<<blocks(GRAPHS * NCLS, TB), TB, 0, stream>>>(f2, fc3w, fc3b, out);
}